// MultiSourceSequenceGenerator_37701222924799
// MI455X (gfx1250) — compile-verified
//
#include <hip/hip_runtime.h>
#include <math.h>

// ---- problem constants (mirror reference) ----
#define BB      32
#define KK      4
#define VV      32000
#define DD      512
#define MAXLEN  64
#define PAD_T   1
#define EOS_T   2
#define UNK_T   3
#define MINLEN  1
#define WORD_R  0.5f
#define UNK_R   (-0.5f)
#define NEGV    (-1e9f)
#define NROW    (BB*KK)          // 128
#define NCHUNK  64               // top-k stage-1 chunks per batch
#define CHSZ    ((KK*VV)/NCHUNK) // 2000
#define NWAVE   4                // waves per k_logits block

typedef __attribute__((ext_vector_type(2))) float        v2f;
typedef __attribute__((ext_vector_type(8))) float        v8f;
typedef __attribute__((ext_vector_type(4))) unsigned int v4u;
typedef __attribute__((ext_vector_type(8))) int          v8i;
typedef __attribute__((ext_vector_type(4))) int          v4i;

// ---------------- init ----------------
__global__ void k_init(int* tokens, float* scores, int* finished,
                       float* lengths, int* prev) {
  int i = blockIdx.x * blockDim.x + threadIdx.x;
  if (i < NROW * (MAXLEN + 1)) tokens[i] = ((i % (MAXLEN + 1)) == 0) ? EOS_T : PAD_T;
  if (i < NROW) {
    scores[i]   = ((i % KK) == 0) ? 0.0f : NEGV;
    finished[i] = 0;
    lengths[i]  = 0.0f;
    prev[i]     = EOS_T;
  }
}

// ---------------- h = tanh(embed[prev] @ W_h + ctx) ----------------
// one wave (32 threads) per 16x16 output tile; K-loop of 16x16x4 f32 WMMA
__global__ void k_h(const float* __restrict__ embed, const float* __restrict__ Wh,
                    const float* __restrict__ ctx, const int* __restrict__ prev,
                    float* __restrict__ h) {
  const int lane = threadIdx.x;
  const int half = lane >> 4;
  const int l    = lane & 15;
  const int r0   = blockIdx.y * 16;   // over NROW=128
  const int c0   = blockIdx.x * 16;   // over DD=512
  const int pr   = prev[r0 + l];
  const float* __restrict__ arow = embed + (size_t)pr * DD;

  v8f acc = {};
  #pragma unroll 4
  for (int k0 = 0; k0 < DD; k0 += 4) {
    const int ka = k0 + 2 * half;
    v2f a; a.x = arow[ka];               a.y = arow[ka + 1];
    v2f b; b.x = Wh[ka * DD + c0 + l];   b.y = Wh[(ka + 1) * DD + c0 + l];
    acc = __builtin_amdgcn_wmma_f32_16x16x4_f32(false, a, false, b,
                                                (short)0, acc, false, false);
  }
  #pragma unroll
  for (int g = 0; g < 8; ++g) {
    const int r = r0 + g + 8 * half;
    const int c = c0 + l;
    h[r * DD + c] = tanhf(acc[g] + ctx[(r / KK) * DD + c]);
  }
}

// ---------------- logits = h @ W_out ----------------
// 4 waves / block share one 16x512 A-tile staged into LDS by the Tensor Data
// Mover (padded: +1 DWORD per 256 so A reads are bank-conflict-free).
// A element n lives at LDS dword n + n/256  ->  idx = l*514 + ka + (ka>>8).
__global__ void k_logits(const float* __restrict__ h, const float* __restrict__ Wout,
                         float* __restrict__ logits) {
  __shared__ float lds_a[16 * 514];   // 8224 dwords >= 8192 + 31 pad
  const int tid  = threadIdx.x;
  const int wave = tid >> 5;
  const int lane = tid & 31;
  const int half = lane >> 4;
  const int l    = lane & 15;
  const int r0   = blockIdx.y * 16;                    // over NROW
  const int c    = (blockIdx.x * NWAVE + wave) * 16;   // this wave's col tile

#if __has_builtin(__builtin_amdgcn_tensor_load_to_lds)
  if (wave == 0) {
    const unsigned lds_base = (unsigned)(uintptr_t)(&lds_a[0]);
    const unsigned long long ga =
        (unsigned long long)(uintptr_t)(h + (size_t)r0 * DD);
    v4u g0;
    g0.x = 1u;                                     // count=1 (valid user D#)
    g0.y = lds_base;                               // lds_addr (bytes)
    g0.z = (unsigned)ga;                           // global_addr[31:0]
    g0.w = (unsigned)(ga >> 32) | (2u << 30);      // global_addr[56:32] | type=2
    v8i g1 = {};
    g1[0] = (2 << 16) | (1 << 20) | (7 << 22);     // data_size=4B, pad_en, ivl=256dw (pad 1dw)
    g1[1] = (int)(8192u << 16);                    // tensor_dim0 = 8192 (lo16 @ bits 63:48)
    g1[2] = (1 << 16);                             // tensor_dim1 = 1
    g1[3] = (int)(8192u << 16);                    // tile_dim0 = 8192
    g1[4] = 0;                                     // tile_dim1 unused (1-D tile)
    g1[5] = 8192;                                  // tensor_dim0_stride
    v4i gz = {};
#if __clang_major__ >= 23
    v8i gz8 = {};
    __builtin_amdgcn_tensor_load_to_lds(g0, g1, gz, gz, gz8, 0);
#else
    __builtin_amdgcn_tensor_load_to_lds(g0, g1, gz, gz, 0);
#endif
    __builtin_amdgcn_s_wait_tensorcnt(0);
  }
  __syncthreads();
#else
  // fallback: cooperative staging with the same padded layout
  for (int i = tid; i < 16 * DD; i += 32 * NWAVE) {
    const int col = i & (DD - 1);
    lds_a[(i >> 9) * 514 + col + (col >> 8)] = h[(size_t)r0 * DD + i];
  }
  __syncthreads();
#endif

  v8f acc = {};
  #pragma unroll 4
  for (int k0 = 0; k0 < DD; k0 += 4) {
    const int ka = k0 + 2 * half;
    const int ai = l * 514 + ka + (ka >> 8);
    v2f a; a.x = lds_a[ai];                       a.y = lds_a[ai + 1];
    v2f b; b.x = Wout[(size_t)ka * VV + c + l];   b.y = Wout[(size_t)(ka + 1) * VV + c + l];
    acc = __builtin_amdgcn_wmma_f32_16x16x4_f32(false, a, false, b,
                                                (short)0, acc, false, false);
  }
  #pragma unroll
  for (int g = 0; g < 8; ++g) {
    logits[(size_t)(r0 + g + 8 * half) * VV + c + l] = acc[g];
  }
}

// ---------------- per-row max and log-sum-exp ----------------
__global__ void k_lse(const float* __restrict__ logits,
                      float* __restrict__ rowmax, float* __restrict__ rowlse) {
  const int r = blockIdx.x, tid = threadIdx.x;
  __shared__ float red[256];
  const float* __restrict__ row = logits + (size_t)r * VV;
  float m = -INFINITY;
  for (int i = tid; i < VV; i += 256) m = fmaxf(m, row[i]);
  red[tid] = m; __syncthreads();
  for (int s = 128; s > 0; s >>= 1) {
    if (tid < s) red[tid] = fmaxf(red[tid], red[tid + s]);
    __syncthreads();
  }
  m = red[0]; __syncthreads();
  float sum = 0.0f;
  for (int i = tid; i < VV; i += 256) sum += expf(row[i] - m);
  red[tid] = sum; __syncthreads();
  for (int s = 128; s > 0; s >>= 1) {
    if (tid < s) red[tid] += red[tid + s];
    __syncthreads();
  }
  if (tid == 0) { rowmax[r] = m; rowlse[r] = logf(red[0]); }
}

// top-4 insertion, tie-break: value desc, index asc (jax.lax.top_k order)
__device__ __forceinline__ void ins4(float bv[4], int bi[4], float v, int i) {
  #pragma unroll
  for (int s = 0; s < 4; ++s) {
    bool better = (v > bv[s]) || (v == bv[s] && i < bi[s]);
    if (better) { float tv = bv[s]; int ti = bi[s]; bv[s] = v; bi[s] = i; v = tv; i = ti; }
  }
}

// ---------------- top-k stage 1: per-(batch, chunk) local top-4 ----------------
__global__ void k_topk1(const float* __restrict__ logits, const float* __restrict__ rowmax,
                        const float* __restrict__ rowlse, const float* __restrict__ scores,
                        const int* __restrict__ finished, int t,
                        float* __restrict__ pv, int* __restrict__ pi) {
  const int b = blockIdx.y, chunk = blockIdx.x, tid = threadIdx.x;
  const int start = chunk * CHSZ, end = start + CHSZ;
  float bv[4] = {-INFINITY, -INFINITY, -INFINITY, -INFINITY};
  int   bi[4] = {0x7fffffff, 0x7fffffff, 0x7fffffff, 0x7fffffff};
  for (int i = start + tid; i < end; i += 256) {
    const int k = i / VV;
    const int v = i - k * VV;
    const int r = b * KK + k;
    const float sc = scores[r];
    float val;
    if (finished[r]) {
      val = sc + ((v == PAD_T) ? 0.0f : NEGV);
    } else {
      float lp = logits[(size_t)r * VV + v] - rowmax[r] - rowlse[r];
      lp += WORD_R;
      if (v == EOS_T) { lp -= WORD_R; if (t < MINLEN) lp = NEGV; }
      if (v == UNK_T) lp += UNK_R;
      if (v == PAD_T) lp = NEGV;
      val = sc + lp;
    }
    ins4(bv, bi, val, i);
  }
  __shared__ float sv[256][4];
  __shared__ int   si[256][4];
  #pragma unroll
  for (int s = 0; s < 4; ++s) { sv[tid][s] = bv[s]; si[tid][s] = bi[s]; }
  __syncthreads();
  for (int s = 128; s > 0; s >>= 1) {
    if (tid < s) {
      float mv[4]; int mi[4];
      #pragma unroll
      for (int j = 0; j < 4; ++j) { mv[j] = sv[tid][j]; mi[j] = si[tid][j]; }
      #pragma unroll
      for (int j = 0; j < 4; ++j) ins4(mv, mi, sv[tid + s][j], si[tid + s][j]);
      #pragma unroll
      for (int j = 0; j < 4; ++j) { sv[tid][j] = mv[j]; si[tid][j] = mi[j]; }
    }
    __syncthreads();
  }
  if (tid == 0) {
    const int base = (b * NCHUNK + chunk) * 4;
    #pragma unroll
    for (int j = 0; j < 4; ++j) { pv[base + j] = sv[0][j]; pi[base + j] = si[0][j]; }
  }
}

// ---------------- top-k stage 2: reduce NCHUNK*4 partials per batch ----------------
__global__ void k_topk2(const float* __restrict__ pv, const int* __restrict__ pi,
                        float* __restrict__ top_sc, int* __restrict__ beam_ix,
                        int* __restrict__ tok_new) {
  const int b = blockIdx.x, tid = threadIdx.x;  // 256 == NCHUNK*4 entries
  __shared__ float sv[256][4];
  __shared__ int   si[256][4];
  sv[tid][0] = pv[b * NCHUNK * 4 + tid];
  si[tid][0] = pi[b * NCHUNK * 4 + tid];
  #pragma unroll
  for (int s = 1; s < 4; ++s) { sv[tid][s] = -INFINITY; si[tid][s] = 0x7fffffff; }
  __syncthreads();
  for (int s = 128; s > 0; s >>= 1) {
    if (tid < s) {
      float mv[4]; int mi[4];
      #pragma unroll
      for (int j = 0; j < 4; ++j) { mv[j] = sv[tid][j]; mi[j] = si[tid][j]; }
      #pragma unroll
      for (int j = 0; j < 4; ++j) ins4(mv, mi, sv[tid + s][j], si[tid + s][j]);
      #pragma unroll
      for (int j = 0; j < 4; ++j) { sv[tid][j] = mv[j]; si[tid][j] = mi[j]; }
    }
    __syncthreads();
  }
  if (tid < 4) {
    const int idx  = si[0][tid];
    const int beam = idx / VV;
    top_sc [b * KK + tid] = sv[0][tid];
    beam_ix[b * KK + tid] = beam;
    tok_new[b * KK + tid] = idx - beam * VV;
  }
}

// ---------------- beam-state reorder + append ----------------
__global__ void k_update(int t, const float* __restrict__ top_sc,
                         const int* __restrict__ beam_ix, const int* __restrict__ tok_new,
                         int* tokens, float* scores, int* finished,
                         float* lengths, int* prev) {
  const int b = blockIdx.x, tid = threadIdx.x;
  __shared__ int   tkold[KK][MAXLEN + 1];
  __shared__ int   finold[KK];
  __shared__ float lenold[KK];
  for (int i = tid; i < KK * (MAXLEN + 1); i += 256) {
    tkold[i / (MAXLEN + 1)][i % (MAXLEN + 1)] = tokens[b * KK * (MAXLEN + 1) + i];
  }
  if (tid < KK) { finold[tid] = finished[b * KK + tid]; lenold[tid] = lengths[b * KK + tid]; }
  __syncthreads();
  for (int i = tid; i < KK * (MAXLEN + 1); i += 256) {
    const int k = i / (MAXLEN + 1), p = i % (MAXLEN + 1);
    const int src = beam_ix[b * KK + k];
    int val = tkold[src][p];
    if (p == t + 1) val = tok_new[b * KK + k];
    tokens[(b * KK + k) * (MAXLEN + 1) + p] = val;
  }
  if (tid < KK) {
    const int k = tid;
    const int src = beam_ix[b * KK + k];
    const int tk  = tok_new[b * KK + k];
    const int fin = finold[src];
    lengths [b * KK + k] = lenold[src] + (fin ? 0.0f : 1.0f);
    finished[b * KK + k] = fin | (tk == EOS_T);
    scores  [b * KK + k] = top_sc[b * KK + k];
    prev    [b * KK + k] = tk;
  }
}

// ---------------- outputs: tokens (int32 bits) + normalized scores ----------------
__global__ void k_final(const int* __restrict__ tokens, const float* __restrict__ scores,
                        const float* __restrict__ lengths, int* out_tok, float* out_f) {
  const int i = blockIdx.x * blockDim.x + threadIdx.x;
  const int NT = NROW * (MAXLEN + 1);
  if (i < NT) out_tok[i] = tokens[i];
  if (i < NROW) out_f[NT + i] = scores[i] / fmaxf(lengths[i], 1.0f);
}

extern "C" void kernel_launch(void* const* d_in, const int* in_sizes, int n_in,
                              void* d_out, int out_size, void* d_ws, size_t ws_size,
                              hipStream_t stream) {
  const float* embed = (const float*)d_in[0];   // [V, D]
  const float* Wh    = (const float*)d_in[1];   // [D, D]
  const float* Wout  = (const float*)d_in[2];   // [D, V]
  const float* ctx   = (const float*)d_in[3];   // [B, D]

  char* p = (char*)d_ws;
  float* h       = (float*)p; p += (size_t)NROW * DD * 4;
  float* logits  = (float*)p; p += (size_t)NROW * VV * 4;
  float* rowmax  = (float*)p; p += NROW * 4;
  float* rowlse  = (float*)p; p += NROW * 4;
  float* scores  = (float*)p; p += NROW * 4;
  float* lengths = (float*)p; p += NROW * 4;
  float* top_sc  = (float*)p; p += NROW * 4;
  float* pv      = (float*)p; p += BB * NCHUNK * 4 * 4;
  int*   pi      = (int*)p;   p += BB * NCHUNK * 4 * 4;
  int*   finished= (int*)p;   p += NROW * 4;
  int*   prev    = (int*)p;   p += NROW * 4;
  int*   beam_ix = (int*)p;   p += NROW * 4;
  int*   tok_new = (int*)p;   p += NROW * 4;
  int*   tokens  = (int*)p;   p += (size_t)NROW * (MAXLEN + 1) * 4;

  k_init<<<dim3(33), 256, 0, stream>>>(tokens, scores, finished, lengths, prev);

  for (int t = 0; t < MAXLEN; ++t) {
    k_h     <<<dim3(DD / 16, NROW / 16), 32, 0, stream>>>(embed, Wh, ctx, prev, h);
    k_logits<<<dim3(VV / (16 * NWAVE), NROW / 16), 32 * NWAVE, 0, stream>>>(h, Wout, logits);
    k_lse   <<<dim3(NROW), 256, 0, stream>>>(logits, rowmax, rowlse);
    k_topk1 <<<dim3(NCHUNK, BB), 256, 0, stream>>>(logits, rowmax, rowlse, scores,
                                                   finished, t, pv, pi);
    k_topk2 <<<dim3(BB), 256, 0, stream>>>(pv, pi, top_sc, beam_ix, tok_new);
    k_update<<<dim3(BB), 256, 0, stream>>>(t, top_sc, beam_ix, tok_new,
                                           tokens, scores, finished, lengths, prev);
  }

  k_final<<<dim3(33), 256, 0, stream>>>(tokens, scores, lengths,
                                        (int*)d_out, (float*)d_out);
}